// pose_estimate_loss_batch_18279380811824
// MI455X (gfx1250) — compile-verified
//
#include <hip/hip_runtime.h>
#include <hip/hip_bf16.h>

// ---------------------------------------------------------------------------
// Problem constants (match the reference): B=8, L=W=H=128, N=16384
//   total points = 8*16384 = 131072 = 512 blocks * 256 threads exactly,
//   so EXEC is all-ones everywhere (required for WMMA).
// ---------------------------------------------------------------------------
#define TOTAL_PTS  131072
#define NBLOCKS    512
#define TPB        256
#define GRID_DIM   128           // L == W == H
#define DIM_MAX_F  127.0f

typedef __attribute__((ext_vector_type(2))) float v2f;
typedef __attribute__((ext_vector_type(8))) float v8f;

// ---------------------------------------------------------------------------
// Wave32 sum-reduction using V_WMMA_F32_16X16X4_F32 (full f32 precision).
//   A (16x4): lane m (m<16) vgpr0 holds A[m][0] = val_m, lanes 16..31 vgpr0
//             hold A[m][2] = val_{m+16}; vgpr1 (K=1,3) set to 0.
//   B (4x16): all ones  =>  D[m][n] = sum_k A[m][k] = val_m + val_{m+16}.
//   D layout: lane l (l<16) holds D[0..7][l] in its 8 vgprs, lane l>=16 holds
//             D[8..15][l-16]. Summing the 8 regs per lane then one
//             shfl_xor(16) yields the full 32-lane sum in every lane.
// ---------------------------------------------------------------------------
__device__ __forceinline__ float wave_sum_wmma(float v) {
    v2f a; a[0] = v;    a[1] = 0.0f;
    v2f b; b[0] = 1.0f; b[1] = 1.0f;
    v8f c = {};
    v8f d = __builtin_amdgcn_wmma_f32_16x16x4_f32(
        /*neg_a=*/false, a, /*neg_b=*/false, b,
        /*c_mod=*/(short)0, c, /*reuse_a=*/false, /*reuse_b=*/false);
    float s = d[0] + d[1] + d[2] + d[3] + d[4] + d[5] + d[6] + d[7];
    s += __shfl_xor(s, 16, 32);
    return s;
}

__global__ __launch_bounds__(TPB)
void trilerp_huber_kernel(const float* __restrict__ tsdf,
                          const float* __restrict__ pts,
                          const float* __restrict__ unit,
                          float* __restrict__ partials) {
    const int gid   = blockIdx.x * TPB + threadIdx.x;   // 0..131071, exact fit
    const int batch = gid >> 14;                        // / 16384

    // grid_unit (uniform, 3 floats)
    const float ux = unit[0];
    const float uy = unit[1];
    const float uz = unit[2];

    // point
    const float* p = pts + 3 * gid;
    const float qx = p[0] / ux;
    const float qy = p[1] / uy;
    const float qz = p[2] / uz;

    const float fx = floorf(qx);
    const float fy = floorf(qy);
    const float fz = floorf(qz);

    // t = (local+1)/2 simplifies exactly to q - floor(q)
    const float tx = qx - fx;
    const float ty = qy - fy;
    const float tz = qz - fz;

    const int x0 = (int)fminf(fmaxf(fx,        0.0f), DIM_MAX_F);
    const int x1 = (int)fminf(fmaxf(fx + 1.0f, 0.0f), DIM_MAX_F);
    const int y0 = (int)fminf(fmaxf(fy,        0.0f), DIM_MAX_F);
    const int y1 = (int)fminf(fmaxf(fy + 1.0f, 0.0f), DIM_MAX_F);
    const int z0 = (int)fminf(fmaxf(fz,        0.0f), DIM_MAX_F);
    const int z1 = (int)fminf(fmaxf(fz + 1.0f, 0.0f), DIM_MAX_F);

    // 8 independent gathers; grid fits in L2 (64 MB < 192 MB)
    const float* gb = tsdf + ((size_t)batch << 21);     // *128*128*128
    const int rx0 = x0 << 14, rx1 = x1 << 14;           // *128*128
    const int ry0 = y0 << 7,  ry1 = y1 << 7;            // *128

    const float c000 = gb[rx0 + ry0 + z0];
    const float c001 = gb[rx0 + ry0 + z1];
    const float c010 = gb[rx0 + ry1 + z0];
    const float c011 = gb[rx0 + ry1 + z1];
    const float c100 = gb[rx1 + ry0 + z0];
    const float c101 = gb[rx1 + ry0 + z1];
    const float c110 = gb[rx1 + ry1 + z0];
    const float c111 = gb[rx1 + ry1 + z1];

    const float sx0 = 1.0f - tx, sy0 = 1.0f - ty, sz0 = 1.0f - tz;

    // bit==1 -> idx_max & t ; bit==0 -> idx_min & (1-t)
    float sdf = c000 * (sx0 * sy0 * sz0)
              + c001 * (sx0 * sy0 * tz )
              + c010 * (sx0 * ty  * sz0)
              + c011 * (sx0 * ty  * tz )
              + c100 * (tx  * sy0 * sz0)
              + c101 * (tx  * sy0 * tz )
              + c110 * (tx  * ty  * sz0)
              + c111 * (tx  * ty  * tz );

    // smooth-L1 / Huber
    const float av = fabsf(sdf);
    const float hv = (av < 1.0f) ? 0.5f * sdf * sdf : av - 0.5f;

    // --- reduction: wave (WMMA) -> block (LDS) -> per-block partial -------
    const float wsum = wave_sum_wmma(hv);

    __shared__ float lds[TPB / 32];
    const int wave = threadIdx.x >> 5;
    const int lane = threadIdx.x & 31;
    if (lane == 0) lds[wave] = wsum;
    __syncthreads();

    if (wave == 0) {
        float s = (lane < (TPB / 32)) ? lds[lane] : 0.0f;
        s += __shfl_xor(s, 1, 32);
        s += __shfl_xor(s, 2, 32);
        s += __shfl_xor(s, 4, 32);
        if (lane == 0) partials[blockIdx.x] = s;
    }
}

__global__ __launch_bounds__(TPB)
void finalize_kernel(const float* __restrict__ partials,
                     float* __restrict__ out) {
    const int t = threadIdx.x;
    float s = partials[t] + partials[t + TPB];          // 512 partials

    s += __shfl_xor(s, 1, 32);
    s += __shfl_xor(s, 2, 32);
    s += __shfl_xor(s, 4, 32);
    s += __shfl_xor(s, 8, 32);
    s += __shfl_xor(s, 16, 32);

    __shared__ float lds[TPB / 32];
    const int wave = t >> 5;
    const int lane = t & 31;
    if (lane == 0) lds[wave] = s;
    __syncthreads();

    if (wave == 0) {
        float v = (lane < (TPB / 32)) ? lds[lane] : 0.0f;
        v += __shfl_xor(v, 1, 32);
        v += __shfl_xor(v, 2, 32);
        v += __shfl_xor(v, 4, 32);
        if (lane == 0) out[0] = v * (1.0f / (float)TOTAL_PTS);
    }
}

extern "C" void kernel_launch(void* const* d_in, const int* in_sizes, int n_in,
                              void* d_out, int out_size, void* d_ws, size_t ws_size,
                              hipStream_t stream) {
    const float* tsdf = (const float*)d_in[0];   // (8,128,128,128) f32
    const float* pts  = (const float*)d_in[1];   // (8,16384,3)     f32
    const float* unit = (const float*)d_in[2];   // (3,)            f32
    float*       out  = (float*)d_out;           // scalar f32
    float*  partials  = (float*)d_ws;            // 512 floats of scratch

    trilerp_huber_kernel<<<NBLOCKS, TPB, 0, stream>>>(tsdf, pts, unit, partials);
    finalize_kernel<<<1, TPB, 0, stream>>>(partials, out);
}